// Gemma3CacheUpdate_20667382628596
// MI455X (gfx1250) — compile-verified
//
#include <hip/hip_runtime.h>

// KV-cache update for 26 layers (Gemma3-style decode step):
//   out_k[i] = cache_k[i] with row (h, pos, :) <- slice_k[i][h, :]  (4x4096x128, d contig)
//   out_v[i] = cache_v[i] with col (h, :, pos) <- slice_v[i][h, :]  (4x128x4096, s contig)
// 872 MB of streaming traffic, 0 FLOPs -> pure HBM roofline (~37us at 23.3 TB/s).
// Strategy: B128 non-temporal loads/stores, 4 float4s per thread for MLP and
// to amortize the per-block scalar preamble, fused slice patch (cheap VALU).

typedef __attribute__((ext_vector_type(4))) float v4f;

#define NUM_LAYERS 26
#define H 4
#define D 128
#define S 4096
#define CACHE_ELEMS (H * S * D)        // 2,097,152 floats = 8 MB
#define CACHE_V4    (CACHE_ELEMS / 4)  // 524,288 float4s
#define UNROLL 4
#define BLOCK 256
#define V4_PER_BLOCK (BLOCK * UNROLL)  // 1024 float4s = 16 KB per block

struct Params {
    const v4f*   src[2 * NUM_LAYERS];  // even = K cache, odd = V cache (per layer)
    const float* sl [2 * NUM_LAYERS];  // matching 512-float slices (h*128 + d)
    const int*   pos;
    v4f*         out;                  // 52 caches concatenated: k0,v0,k1,v1,...
};

__global__ __launch_bounds__(BLOCK) void kv_cache_update(Params p) {
    const int cidx = blockIdx.y;                               // 0..51, uniform
    const int base = blockIdx.x * V4_PER_BLOCK + threadIdx.x;  // float4 index, stride BLOCK
    const int pos  = *p.pos;                                   // uniform scalar load

    const v4f* __restrict__ src = p.src[cidx];
    v4f*       __restrict__ dst = p.out + (size_t)cidx * CACHE_V4;

    // Issue all 4 independent B128 NT loads first (deep MLP, clause-able).
    v4f v[UNROLL];
#pragma unroll
    for (int k = 0; k < UNROLL; ++k)
        v[k] = __builtin_nontemporal_load(&src[base + k * BLOCK]);

    if ((cidx & 1) == 0) {
        // K cache: (h, s, d), d contiguous (128). float4 = 4 consecutive d;
        // whole 128-wide row replaced when s == pos.
        const v4f* sl4 = (const v4f*)p.sl[cidx];
#pragma unroll
        for (int k = 0; k < UNROLL; ++k) {
            const int e0 = (base + k * BLOCK) << 2;
            const int d  = e0 & (D - 1);
            const int s  = (e0 >> 7) & (S - 1);
            const int h  = e0 >> 19;                 // / (128*4096)
            if (s == pos)
                v[k] = sl4[(h * D + d) >> 2];
        }
    } else {
        // V cache: (h, d, s), s contiguous (4096). float4 covers s0..s0+3;
        // patch single lane-component when pos falls inside the chunk.
        const float* sl = p.sl[cidx];
#pragma unroll
        for (int k = 0; k < UNROLL; ++k) {
            const int e0 = (base + k * BLOCK) << 2;
            const int s0 = e0 & (S - 1);
            const int d  = (e0 >> 12) & (D - 1);
            const int h  = e0 >> 19;
            const int r  = pos - s0;
            if ((unsigned)r < 4u) {
                const float x = sl[h * D + d];
                if      (r == 0) v[k].x = x;
                else if (r == 1) v[k].y = x;
                else if (r == 2) v[k].z = x;
                else             v[k].w = x;
            }
        }
    }

    // 4 independent B128 NT streaming stores.
#pragma unroll
    for (int k = 0; k < UNROLL; ++k)
        __builtin_nontemporal_store(v[k], &dst[base + k * BLOCK]);
}

extern "C" void kernel_launch(void* const* d_in, const int* in_sizes, int n_in,
                              void* d_out, int out_size, void* d_ws, size_t ws_size,
                              hipStream_t stream) {
    (void)in_sizes; (void)n_in; (void)out_size; (void)d_ws; (void)ws_size;

    Params p;
    p.pos = (const int*)d_in[0];
    p.out = (v4f*)d_out;
    for (int i = 0; i < NUM_LAYERS; ++i) {
        // setup_inputs order per layer: cache_k, slice_k, cache_v, slice_v
        p.src[2 * i + 0] = (const v4f*)  d_in[1 + 4 * i + 0];
        p.sl [2 * i + 0] = (const float*)d_in[1 + 4 * i + 1];
        p.src[2 * i + 1] = (const v4f*)  d_in[1 + 4 * i + 2];
        p.sl [2 * i + 1] = (const float*)d_in[1 + 4 * i + 3];
    }

    dim3 grid(CACHE_V4 / V4_PER_BLOCK, 2 * NUM_LAYERS);  // (512, 52)
    kv_cache_update<<<grid, BLOCK, 0, stream>>>(p);
}